// CRF_16741782520291
// MI455X (gfx1250) — compile-verified
//
#include <hip/hip_runtime.h>
#include <hip/hip_bf16.h>
#include <math.h>

// ---------------------------------------------------------------------------
// CRF mean log-likelihood for MI455X (gfx1250, wave32, WMMA).
//
// Inputs: emissions f32[1024,1024,21], tags i32[1024,1024], mask u8[1024,1024],
//         start f32[21], end f32[21], trans f32[21,21].  Output: 1 f32.
//
// Denominator (forward algorithm, sequential in L), log2 domain:
//   beta = alpha*log2(e);  u_i = exp2(beta_i - m);  v = u * exp(trans)  (GEMM)
//   beta'_j = log2(v_j) + m + e_j*log2(e)
// exp(trans) is constant (IMP -> exactly 0) so each step is a
// (16 batches x 24)x(24 x 32) fp32 GEMM -> 12 chained V_WMMA_F32_16X16X4_F32.
// Per-batch max m recomputed each step from LDS alpha (streaming softmax;
// mask-consistent by construction). K-pad columns hold -inf so exp2 gives
// exact zeros with no predicates. Raw v_exp_f32/v_log_f32 via builtins.
// ---------------------------------------------------------------------------

#define TT 21      // number of tags
#define LL 1024    // sequence length
#define BB 1024    // batch
#define AS 26      // LDS row stride: even (ds_load_b64 pairs), conflict-free
#define PF 8       // emission prefetch distance (time steps)

#define LOG2E 1.44269504088896340736f
#define LN2   0.69314718055994530942f

typedef float v2f __attribute__((ext_vector_type(2)));
typedef float v8f __attribute__((ext_vector_type(8)));

static __device__ __forceinline__ float fast_exp2(float x) {
#if defined(__HIP_DEVICE_COMPILE__) && __has_builtin(__builtin_amdgcn_exp2f)
    return __builtin_amdgcn_exp2f(x);   // raw v_exp_f32
#else
    return exp2f(x);
#endif
}
static __device__ __forceinline__ float fast_log2(float x) {
#if defined(__HIP_DEVICE_COMPILE__) && __has_builtin(__builtin_amdgcn_logf)
    return __builtin_amdgcn_logf(x);    // raw v_log_f32
#else
    return log2f(x);
#endif
}

// ------------------------- numerator (path score) --------------------------
__global__ __launch_bounds__(32)
void crf_score_kernel(const float* __restrict__ emis,
                      const int* __restrict__ tags,
                      const unsigned char* __restrict__ maskp,
                      const float* __restrict__ startv,
                      const float* __restrict__ endv,
                      const float* __restrict__ trans,
                      float* __restrict__ wsScore,
                      int* __restrict__ wsCnt) {
    const int b = blockIdx.x;
    const int l = threadIdx.x;
    const int* tg = tags + (size_t)b * LL;
    const unsigned char* mk = maskp + (size_t)b * LL;
    const float* em = emis + (size_t)b * LL * TT;

    float part = 0.f;   // masked sum of trans + emission terms, t >= 1
    float s0   = 0.f;   // unmasked start + emission at t == 0 (ref semantics)
    int   cnt  = 0;     // mask count (for last_idx and global average)

    for (int t = l; t < LL; t += 32) {
        int   tgt = tg[t];
        int   m   = mk[t];
        float e   = em[(size_t)t * TT + tgt];
        if (t == 0) {
            s0 = startv[tgt] + e;
        } else if (m) {
            part += trans[tg[t - 1] * TT + tgt] + e;
        }
        cnt += m ? 1 : 0;
    }
    #pragma unroll
    for (int o = 16; o > 0; o >>= 1) {
        part += __shfl_xor(part, o, 32);
        s0   += __shfl_xor(s0,   o, 32);
        cnt  += __shfl_xor(cnt,  o, 32);
    }
    if (l == 0) {
        int last = cnt - 1;
        if (last < 0) last = 0;
        wsScore[b] = s0 + part + endv[tg[last]];
        wsCnt[b]   = cnt;
    }
}

// ---------------------- denominator (forward / WMMA) -----------------------
__global__ __launch_bounds__(32)
void crf_denom_kernel(const float* __restrict__ emis,
                      const unsigned char* __restrict__ maskp,
                      const float* __restrict__ startv,
                      const float* __restrict__ endv,
                      const float* __restrict__ trans,
                      float* __restrict__ wsDenom) {
    __shared__ float shA[16 * AS];  // log2-domain alpha, 16 batches x 21(+pad)

    const int l     = threadIdx.x;
    const int half  = l >> 4;       // 0: lanes 0-15, 1: lanes 16-31
    const int jj    = l & 15;
    const int bbase = blockIdx.x * 16;

    // Constant B-fragments: exp(trans) (natural!), zero-padded to 24x32.
    // f32 B layout mirrors documented A layout: element e of chunk c holds
    // row K = 4c + e + 2*half, column N = lane&15 (+16 per tile).
    v2f bf[2][6];
    #pragma unroll
    for (int tile = 0; tile < 2; ++tile) {
        int j = jj + 16 * tile;
        #pragma unroll
        for (int c = 0; c < 6; ++c) {
            int k0 = 4 * c + 2 * half;
            bf[tile][c].x = (k0     < TT && j < TT) ? __expf(trans[k0       * TT + j]) : 0.f;
            bf[tile][c].y = (k0 + 1 < TT && j < TT) ? __expf(trans[(k0 + 1) * TT + j]) : 0.f;
        }
    }

    // beta0 = (start + emissions[:,0,:]) * log2e; K-pad columns = -inf so
    // exp2(pad - m) == 0 exactly -> branch-free phase A.
    if (l < 16) {
        size_t base = (size_t)(bbase + l) * LL * TT;
        #pragma unroll
        for (int j = 0; j < TT; ++j)
            shA[l * AS + j] = (startv[j] + emis[base + j]) * LOG2E;
        shA[l * AS + 21] = -INFINITY;
        shA[l * AS + 22] = -INFINITY;
        shA[l * AS + 23] = -INFINITY;
    }
    __syncthreads();

    // Mask byte: every lane loads batch (bbase+jj) -> halves broadcast the
    // same address (no extra cache lines, no EXEC branch).
    const unsigned char* mrow = maskp + (size_t)(bbase + jj) * LL;
    const bool hi = (jj + 16 < TT);
    const int  j1 = hi ? jj + 16 : jj;   // clamped second-emission column

    for (int t = 1; t < LL; ++t) {
        __syncthreads();  // fence previous step's LDS writes vs this read phase

        // Prefetch emission rows for step t+PF (84 B per batch row; the two
        // lane-halves cover both cache lines). -> global_prefetch_b8
        {
            int tp = t + PF < LL ? t + PF : LL - 1;
            const float* p = emis + ((size_t)(bbase + jj) * LL + tp) * TT;
            __builtin_prefetch(p + half * 16, 0, 0);
        }

        // ---- Phase A: load beta, fresh per-batch max, u = exp2(beta - m) ---
        // A 16x4 f32 layout: lane = batch row (l&15); K = 2*half + elem (+4c).
        float x0[6], x1[6];
        #pragma unroll
        for (int c = 0; c < 6; ++c) {
            int k0 = 4 * c + 2 * half;
            x0[c] = shA[jj * AS + k0];
            x1[c] = shA[jj * AS + k0 + 1];
        }
        // The 12 values per lane + 12 in the other half tile a batch's 21
        // states (+3 -inf pads): 11 fmax + one cross-half xor = exact max.
        float mA = fmaxf(x0[0], x1[0]);
        #pragma unroll
        for (int c = 1; c < 6; ++c) mA = fmaxf(mA, fmaxf(x0[c], x1[c]));
        mA = fmaxf(mA, __shfl_xor(mA, 16, 32));

        v2f a[6];
        #pragma unroll
        for (int c = 0; c < 6; ++c) {
            a[c].x = fast_exp2(x0[c] - mA);
            a[c].y = fast_exp2(x1[c] - mA);
        }

        // ---- 12 chained fp32 WMMAs: v = u * exp(trans) ----
        v8f acc0 = {};
        v8f acc1 = {};
        #pragma unroll
        for (int c = 0; c < 6; ++c) {
            acc0 = __builtin_amdgcn_wmma_f32_16x16x4_f32(
                false, a[c], false, bf[0][c], (short)0, acc0, false, false);
            acc1 = __builtin_amdgcn_wmma_f32_16x16x4_f32(
                false, a[c], false, bf[1][c], (short)0, acc1, false, false);
        }

        // ---- Phase C: beta' = log2(v) + m + e*log2e; masked writeback ----
        int mByte = (int)mrow[t];
        unsigned bal = (unsigned)__ballot(mByte != 0);  // bits 0-15 = batches
        #pragma unroll
        for (int r = 0; r < 8; ++r) {
            int   b_r  = r + 8 * half;               // batch row in C layout
            int   mb   = (bal >> b_r) & 1;           // mask[b][t]
            float mOld = __shfl(mA, b_r, 32);        // max used to build u
            size_t eb  = ((size_t)(bbase + b_r) * LL + t) * TT;
            float e0   = emis[eb + jj];
            float e1   = emis[eb + j1];              // clamped; hi-guarded use
            float n0   = fmaf(e0, LOG2E, fast_log2(acc0[r]) + mOld);
            float n1   = fmaf(e1, LOG2E, fast_log2(acc1[r]) + mOld);
            if (mb) {  // where(mask, nxt, alpha): skip writes to keep old state
                shA[b_r * AS + jj] = n0;
                if (hi) shA[b_r * AS + jj + 16] = n1;
            }
        }
    }
    __syncthreads();

    // denom = logsumexp(alpha + end), done in log2 domain then scaled back.
    if (l < 16) {
        float be[TT];
        float mx = -INFINITY;
        #pragma unroll
        for (int j = 0; j < TT; ++j) {
            be[j] = shA[l * AS + j] + endv[j] * LOG2E;
            mx = fmaxf(mx, be[j]);
        }
        float s = 0.f;
        #pragma unroll
        for (int j = 0; j < TT; ++j)
            s += fast_exp2(be[j] - mx);
        wsDenom[bbase + l] = (fast_log2(s) + mx) * LN2;
    }
}

// --------------------------- final reduction -------------------------------
__global__ __launch_bounds__(256)
void crf_final_kernel(const float* __restrict__ wsScore,
                      const float* __restrict__ wsDenom,
                      const int* __restrict__ wsCnt,
                      float* __restrict__ out) {
    __shared__ float sred[256];
    __shared__ int   cred[256];
    int tid = threadIdx.x;
    float s = 0.f;
    int   c = 0;
    for (int i = tid; i < BB; i += 256) {
        s += wsScore[i] - wsDenom[i];
        c += wsCnt[i];
    }
    sred[tid] = s;
    cred[tid] = c;
    __syncthreads();
    for (int o = 128; o > 0; o >>= 1) {
        if (tid < o) {
            sred[tid] += sred[tid + o];
            cred[tid] += cred[tid + o];
        }
        __syncthreads();
    }
    if (tid == 0) out[0] = sred[0] / (float)cred[0];
}

// ------------------------------- launcher ----------------------------------
extern "C" void kernel_launch(void* const* d_in, const int* in_sizes, int n_in,
                              void* d_out, int out_size, void* d_ws, size_t ws_size,
                              hipStream_t stream) {
    const float*         emis   = (const float*)d_in[0];
    const int*           tags   = (const int*)d_in[1];
    const unsigned char* maskp  = (const unsigned char*)d_in[2];
    const float*         startv = (const float*)d_in[3];
    const float*         endv   = (const float*)d_in[4];
    const float*         trans  = (const float*)d_in[5];
    float*               out    = (float*)d_out;

    float* wsScore = (float*)d_ws;          // [BB]
    float* wsDenom = wsScore + BB;          // [BB]
    int*   wsCnt   = (int*)(wsDenom + BB);  // [BB]   (12 KB total scratch)

    crf_score_kernel<<<BB, 32, 0, stream>>>(emis, tags, maskp, startv, endv,
                                            trans, wsScore, wsCnt);
    crf_denom_kernel<<<BB / 16, 32, 0, stream>>>(emis, maskp, startv, endv,
                                                 trans, wsDenom);
    crf_final_kernel<<<1, 256, 0, stream>>>(wsScore, wsDenom, wsCnt, out);
}